// MultiModalVAE_50483045597836
// MI455X (gfx1250) — compile-verified
//
#include <hip/hip_runtime.h>
#include <math.h>

#define BB 16384
#define MM 5
#define DD 512
#define LL 64
#define EE 112
#define HH 8
#define DH 14
#define NBLK 4

typedef __attribute__((ext_vector_type(2))) float v2f;
typedef __attribute__((ext_vector_type(8))) float v8f;

__device__ __forceinline__ float gelu_f(float x) {
  return 0.5f * x * (1.0f + erff(x * 0.70710678118654752f));
}

__device__ __forceinline__ v8f wmma4(v2f a, v2f b, v8f c) {
  // V_WMMA_F32_16X16X4_F32 : D = A(16x4) * B(4x16) + C(16x16), fp32
  return __builtin_amdgcn_wmma_f32_16x16x4_f32(false, a, false, b, (short)0, c,
                                               false, false);
}

// ---------------------------------------------------------------------------
// batched weight transpose: Wt[m][n][k] = W[m][k][n]  (N-major copy so B
// fragments become single aligned b64 loads)
// ---------------------------------------------------------------------------
__global__ void k_transpose(const float* __restrict__ W, float* __restrict__ Wt,
                            int K, int N) {
  long base = (long)blockIdx.y * K * N;
  long total = (long)K * N;
  for (long i = (long)blockIdx.x * blockDim.x + threadIdx.x; i < total;
       i += (long)gridDim.x * blockDim.x) {
    int k = (int)(i / N), n = (int)(i % N);
    Wt[base + (long)n * K + k] = W[base + i];
  }
}

// ---------------------------------------------------------------------------
// batched GEMM, transposed-B, 4-way N-register-blocked:
//   C[m] = A[m] (R x K) @ B[m] (K x N) + bias[m],  Bt laid out N x K
// one 16 x 64 block per wave: 1 shared A-frag per k-step feeds 4 WMMAs
// ---------------------------------------------------------------------------
__global__ void k_gemm_bias_t(const float* __restrict__ A, long lda, long abatch,
                              const float* __restrict__ Bt, long bbatch,
                              const float* __restrict__ bias, long biasbatch,
                              float* __restrict__ C, long ldc, long cbatch,
                              int R, int K, int N) {
  int m = blockIdx.y;
  const float* Ab = A + (long)m * abatch;
  const float* Bb = Bt + (long)m * bbatch;
  const float* biasb = bias + (long)m * biasbatch;
  float* Cb = C + (long)m * cbatch;
  int lane = threadIdx.x & 31, l15 = lane & 15, hl = lane >> 4;
  int wid = blockIdx.x * (blockDim.x >> 5) + (threadIdx.x >> 5);
  int wstride = gridDim.x * (blockDim.x >> 5);
  int tilesN = N >> 4;
  int groupsN = (tilesN + NBLK - 1) / NBLK;
  int groups = (R >> 4) * groupsN;
  for (int t = wid; t < groups; t += wstride) {
    int tr = t / groupsN, tg = t % groupsN;
    int tn0 = tg * NBLK;
    int nb = tilesN - tn0; if (nb > NBLK) nb = NBLK;
    const float* arow = Ab + (long)(tr * 16 + l15) * lda + hl * 2;
    const float* brow[NBLK];
    v8f acc[NBLK];
#pragma unroll
    for (int j = 0; j < NBLK; j++) {
      acc[j] = (v8f){};
      int tn = tn0 + (j < nb ? j : 0);
      brow[j] = Bb + (long)(tn * 16 + l15) * K + hl * 2;
    }
    for (int k = 0; k < K; k += 4) {
      v2f a = *(const v2f*)(arow + k);
#pragma unroll
      for (int j = 0; j < NBLK; j++) {
        v2f b = *(const v2f*)(brow[j] + k);
        acc[j] = wmma4(a, b, acc[j]);
      }
    }
#pragma unroll
    for (int j = 0; j < NBLK; j++) {
      if (j < nb) {
        int tn = tn0 + j;
        float bv = biasb[tn * 16 + l15];
        float* cp = Cb + (long)(tr * 16 + hl * 8) * ldc + tn * 16 + l15;
#pragma unroll
        for (int i = 0; i < 8; i++) cp[(long)i * ldc] = acc[j][i] + bv;
      }
    }
  }
}

// same, with per-row qw*mask scaling epilogue (enc slices)
__global__ void k_gemm_enc_t(const float* __restrict__ A, long lda, long abatch,
                             const float* __restrict__ Bt, long bbatch,
                             const float* __restrict__ bias, long biasbatch,
                             const float* __restrict__ qw, const int* __restrict__ pm,
                             float* __restrict__ C, long ldc, long cbatch,
                             int R, int K, int N) {
  int m = blockIdx.y;
  const float* Ab = A + (long)m * abatch;
  const float* Bb = Bt + (long)m * bbatch;
  const float* biasb = bias + (long)m * biasbatch;
  float* Cb = C + (long)m * cbatch;
  int lane = threadIdx.x & 31, l15 = lane & 15, hl = lane >> 4;
  int wid = blockIdx.x * (blockDim.x >> 5) + (threadIdx.x >> 5);
  int wstride = gridDim.x * (blockDim.x >> 5);
  int tilesN = N >> 4;
  int groupsN = (tilesN + NBLK - 1) / NBLK;
  int groups = (R >> 4) * groupsN;
  for (int t = wid; t < groups; t += wstride) {
    int tr = t / groupsN, tg = t % groupsN;
    int tn0 = tg * NBLK;
    int nb = tilesN - tn0; if (nb > NBLK) nb = NBLK;
    const float* arow = Ab + (long)(tr * 16 + l15) * lda + hl * 2;
    const float* brow[NBLK];
    v8f acc[NBLK];
#pragma unroll
    for (int j = 0; j < NBLK; j++) {
      acc[j] = (v8f){};
      int tn = tn0 + (j < nb ? j : 0);
      brow[j] = Bb + (long)(tn * 16 + l15) * K + hl * 2;
    }
    for (int k = 0; k < K; k += 4) {
      v2f a = *(const v2f*)(arow + k);
#pragma unroll
      for (int j = 0; j < NBLK; j++) {
        v2f b = *(const v2f*)(brow[j] + k);
        acc[j] = wmma4(a, b, acc[j]);
      }
    }
#pragma unroll
    for (int j = 0; j < NBLK; j++) {
      if (j < nb) {
        int tn = tn0 + j;
        float bv = biasb[tn * 16 + l15];
        float* cp = Cb + (long)(tr * 16 + hl * 8) * ldc + tn * 16 + l15;
#pragma unroll
        for (int i = 0; i < 8; i++) {
          int r = tr * 16 + hl * 8 + i;  // batch row (b index)
          float sc = pm[r * MM + m] ? qw[r * MM + m] : 0.0f;
          cp[(long)i * ldc] = (acc[j][i] + bv) * sc;
        }
      }
    }
  }
}

// ---------------------------------------------------------------------------
// present-mask counts per modality
// ---------------------------------------------------------------------------
__global__ void k_cnt(const int* __restrict__ pm, float* __restrict__ cnt) {
  int b = blockIdx.x * blockDim.x + threadIdx.x;
  int lane = threadIdx.x & 31;
  float lc[MM];
#pragma unroll
  for (int m = 0; m < MM; m++) lc[m] = (b < BB && pm[b * MM + m]) ? 1.0f : 0.0f;
#pragma unroll
  for (int m = 0; m < MM; m++) {
    float s = lc[m];
    for (int off = 16; off > 0; off >>= 1) s += __shfl_down(s, off, 32);
    if (lane == 0) atomicAdd(&cnt[m], s);
  }
}

// ---------------------------------------------------------------------------
// quality transform: Linear(5->64) -> exact GELU -> Linear(64->5) -> softmax
// ---------------------------------------------------------------------------
__global__ void k_qw(const float* __restrict__ quality, const float* __restrict__ W1,
                     const float* __restrict__ b1, const float* __restrict__ W2,
                     const float* __restrict__ b2, float* __restrict__ qw) {
  int b = blockIdx.x * blockDim.x + threadIdx.x;
  if (b >= BB) return;
  float qr[MM], o[MM];
#pragma unroll
  for (int m = 0; m < MM; m++) { qr[m] = quality[b * MM + m]; o[m] = b2[m]; }
  for (int j = 0; j < 64; j++) {
    float hv = b1[j];
#pragma unroll
    for (int m = 0; m < MM; m++) hv += qr[m] * W1[m * 64 + j];
    float g = gelu_f(hv);
#pragma unroll
    for (int m = 0; m < MM; m++) o[m] += g * W2[j * MM + m];
  }
  float mx = o[0];
#pragma unroll
  for (int m = 1; m < MM; m++) mx = fmaxf(mx, o[m]);
  float den = 0.0f;
#pragma unroll
  for (int m = 0; m < MM; m++) { o[m] = expf(o[m] - mx); den += o[m]; }
  float inv = 1.0f / den;
#pragma unroll
  for (int m = 0; m < MM; m++) qw[b * MM + m] = o[m] * inv;
}

// ---------------------------------------------------------------------------
// masked moments of h over B per (m, channel)
// ---------------------------------------------------------------------------
__global__ void k_stats(const float* __restrict__ h, const int* __restrict__ pm,
                        float* __restrict__ ssum, float* __restrict__ ssq, int ds) {
  int c = blockIdx.x * blockDim.x + threadIdx.x;
  int nch = MM * ds;
  if (c >= nch) return;
  int m = c / ds, e = c % ds;
  int b0 = blockIdx.y * (BB / 64);
  float s = 0.0f, s2 = 0.0f;
  for (int b = b0; b < b0 + BB / 64; b++) {
    if (pm[b * MM + m]) {
      float v = h[((long)b * MM + m) * ds + e];
      s += v;
      s2 += v * v;
    }
  }
  atomicAdd(&ssum[m * 512 + e], s);
  atomicAdd(&ssq[m * 512 + e], s2);
}

// fold BN into affine coeffs: hn = h*aa + cc
__global__ void k_bnfin(const float* __restrict__ ssum, const float* __restrict__ ssq,
                        const float* __restrict__ cnt, const float* __restrict__ g,
                        const float* __restrict__ be, float* __restrict__ aa,
                        float* __restrict__ cc, int ds) {
  int c = blockIdx.x * blockDim.x + threadIdx.x;
  if (c >= MM * ds) return;
  int m = c / ds, e = c % ds;
  float cn = fmaxf(cnt[m], 1.0f);
  float mean = ssum[m * 512 + e] / cn;
  float var = ssq[m * 512 + e] / cn - mean * mean;
  float istd = rsqrtf(var + 1e-5f);
  float av = istd * g[c];
  aa[m * 512 + e] = av;
  cc[m * 512 + e] = be[c] - mean * av;
}

// in-place normalized GELU
__global__ void k_hn(float* __restrict__ h, const float* __restrict__ aa,
                     const float* __restrict__ cc, int ds) {
  long total = (long)BB * MM * ds;
  for (long i = (long)blockIdx.x * blockDim.x + threadIdx.x; i < total;
       i += (long)gridDim.x * blockDim.x) {
    int e = (int)(i % ds);
    int m = (int)((i / ds) % MM);
    h[i] = gelu_f(h[i] * aa[m * 512 + e] + cc[m * 512 + e]);
  }
}

// ---------------------------------------------------------------------------
// masked multihead attention core (M=5, per (b,head) thread)
// ---------------------------------------------------------------------------
__global__ void k_attn(const float* __restrict__ q, const float* __restrict__ k,
                       const float* __restrict__ v, const int* __restrict__ pm,
                       float* __restrict__ att) {
  int idx = blockIdx.x * blockDim.x + threadIdx.x;
  if (idx >= BB * HH) return;
  int b = idx / HH, h = idx % HH, e0 = h * DH;
  const float scale = 0.2672612419124244f;  // 1/sqrt(14)
  for (int mq = 0; mq < MM; mq++) {
    float qr[DH];
#pragma unroll
    for (int j = 0; j < DH; j++) qr[j] = q[((long)b * MM + mq) * EE + e0 + j];
    float sc[MM];
    float mx = -3.0e38f;
#pragma unroll
    for (int mk = 0; mk < MM; mk++) {
      float s = 0.0f;
#pragma unroll
      for (int j = 0; j < DH; j++) s += qr[j] * k[((long)b * MM + mk) * EE + e0 + j];
      s *= scale;
      if (pm[b * MM + mk] == 0) s = -1e9f;
      sc[mk] = s;
      mx = fmaxf(mx, s);
    }
    float den = 0.0f;
#pragma unroll
    for (int mk = 0; mk < MM; mk++) { sc[mk] = expf(sc[mk] - mx); den += sc[mk]; }
    float inv = 1.0f / den;
#pragma unroll
    for (int j = 0; j < DH; j++) {
      float o = 0.0f;
#pragma unroll
      for (int mk = 0; mk < MM; mk++)
        o += sc[mk] * v[((long)b * MM + mk) * EE + e0 + j];
      att[((long)b * MM + mq) * EE + e0 + j] = o * inv;
    }
  }
}

// availability-weighted pooling -> joint
__global__ void k_pool(const float* __restrict__ enc, const float* __restrict__ att2,
                       const int* __restrict__ pm, float* __restrict__ joint) {
  int idx = blockIdx.x * blockDim.x + threadIdx.x;
  if (idx >= BB * EE) return;
  int b = idx / EE, e = idx % EE;
  float na = 0.0f, s = 0.0f;
#pragma unroll
  for (int m = 0; m < MM; m++) {
    if (pm[b * MM + m]) {
      na += 1.0f;
      long r = (long)b * MM + m;
      s += enc[r * EE + e] + 0.5f * att2[r * EE + e];
    }
  }
  joint[(long)b * EE + e] = s / fmaxf(na, 1.0f);
}

// ---------------------------------------------------------------------------
// mu / logvar head: gelu(joint@W1+b1)@W2+b2 fused via LDS (transposed weights)
// block = 128 threads (4 waves), grid = B/16
// ---------------------------------------------------------------------------
__global__ void k_head(const float* __restrict__ joint, const float* __restrict__ W1t,
                       const float* __restrict__ b1, const float* __restrict__ W2t,
                       const float* __restrict__ b2, float* __restrict__ outp) {
  __shared__ float t1[16 * 68];
  int lane = threadIdx.x & 31, l15 = lane & 15, hl = lane >> 4;
  int w = threadIdx.x >> 5;  // 0..3 -> 16-col tile of the 64 hidden cols
  int tr = blockIdx.x;
  // stage 1: t1 = gelu(joint(16xE) @ W1(ExL) + b1)
  {
    const float* arow = joint + (long)(tr * 16 + l15) * EE + hl * 2;
    const float* brow = W1t + (long)(w * 16 + l15) * EE + hl * 2;  // W1t: L x E
    v8f acc = {};
    for (int k = 0; k < EE; k += 4) {
      v2f a = *(const v2f*)(arow + k);
      v2f b = *(const v2f*)(brow + k);
      acc = wmma4(a, b, acc);
    }
    float bv = b1[w * 16 + l15];
#pragma unroll
    for (int i = 0; i < 8; i++)
      t1[(hl * 8 + i) * 68 + w * 16 + l15] = gelu_f(acc[i] + bv);
  }
  __syncthreads();
  // stage 2: out = t1(16xL) @ W2(LxL) + b2
  {
    const float* a2 = &t1[l15 * 68 + hl * 2];
    const float* brow = W2t + (long)(w * 16 + l15) * LL + hl * 2;  // W2t: L x L
    v8f acc = {};
    for (int k = 0; k < LL; k += 4) {
      v2f a = *(const v2f*)(a2 + k);
      v2f b = *(const v2f*)(brow + k);
      acc = wmma4(a, b, acc);
    }
    float bv = b2[w * 16 + l15];
    float* cp = outp + (long)(tr * 16 + hl * 8) * LL + w * 16 + l15;
#pragma unroll
    for (int i = 0; i < 8; i++) cp[(long)i * LL] = acc[i] + bv;
  }
}

// reparameterize + KL partial sum
__global__ void k_z(const float* __restrict__ mu, const float* __restrict__ lv,
                    const float* __restrict__ eps, float* __restrict__ z,
                    float* __restrict__ klacc) {
  int idx = blockIdx.x * blockDim.x + threadIdx.x;
  int lane = threadIdx.x & 31;
  float kl = 0.0f;
  if (idx < BB * LL) {
    float m = mu[idx], l = lv[idx];
    z[idx] = m + eps[idx] * expf(0.5f * l);
    kl = 1.0f + l - m * m - expf(l);
  }
  for (int off = 16; off > 0; off >>= 1) kl += __shfl_down(kl, off, 32);
  if (lane == 0) atomicAdd(klacc, kl);
}

// ---------------------------------------------------------------------------
// fused decoder (transposed weights): d1 = gelu(z@W1+b1) [LDS],
// decoded = d1@W2+b2, imputation + masked recon accumulation.
// block = 256 (8 waves), grid = (B/16, M); k-outer loops share one A-frag
// across each wave's 4 accumulator tiles.
// ---------------------------------------------------------------------------
#define D1LD 516  // 516 % 64 == 4 -> conflict-free fragment reads
__global__ void k_decode(const float* __restrict__ z, const float* __restrict__ W1t,
                         const float* __restrict__ b1, const float* __restrict__ W2t,
                         const float* __restrict__ b2, const float* __restrict__ emb,
                         const int* __restrict__ pm, float* __restrict__ out_imp,
                         float* __restrict__ recon_acc) {
  __shared__ float zs[16 * 68];
  __shared__ float d1[16 * D1LD];
  int m = blockIdx.y, tr = blockIdx.x;
  const float* W1tm = W1t + (long)m * LL * 1024;   // 1024 x 64
  const float* b1m = b1 + (long)m * 1024;
  const float* W2tm = W2t + (long)m * 1024 * DD;   // 512 x 1024
  const float* b2m = b2 + (long)m * DD;
  for (int i = threadIdx.x; i < 16 * LL; i += blockDim.x) {
    int r = i >> 6, c = i & 63;
    zs[r * 68 + c] = z[(long)(tr * 16 + r) * LL + c];
  }
  __syncthreads();
  int lane = threadIdx.x & 31, l15 = lane & 15, hl = lane >> 4;
  int w = threadIdx.x >> 5;  // wave 0..7
  v8f dec[4];
#pragma unroll
  for (int j = 0; j < 4; j++) dec[j] = (v8f){};
  for (int chunk = 0; chunk < 2; chunk++) {
    int ncol0 = chunk * 512;
    // stage 1: this chunk's 16x512 slab of d1 (4 N-tiles per wave, shared A)
    {
      v8f a1[4];
      const float* brow1[4];
#pragma unroll
      for (int j = 0; j < 4; j++) {
        a1[j] = (v8f){};
        int nc = ncol0 + (w + j * 8) * 16;
        brow1[j] = W1tm + (long)(nc + l15) * LL + hl * 2;
      }
      const float* a2 = &zs[l15 * 68 + hl * 2];
      for (int k = 0; k < LL; k += 4) {
        v2f a = *(const v2f*)(a2 + k);
#pragma unroll
        for (int j = 0; j < 4; j++) {
          v2f b = *(const v2f*)(brow1[j] + k);
          a1[j] = wmma4(a, b, a1[j]);
        }
      }
#pragma unroll
      for (int j = 0; j < 4; j++) {
        int tt = w + j * 8;
        float bv = b1m[ncol0 + tt * 16 + l15];
#pragma unroll
        for (int i = 0; i < 8; i++)
          d1[(hl * 8 + i) * D1LD + tt * 16 + l15] = gelu_f(a1[j][i] + bv);
      }
    }
    __syncthreads();
    // stage 2: accumulate decoded over this chunk's K range (shared A-frag)
    {
      const float* brow2[4];
#pragma unroll
      for (int j = 0; j < 4; j++) {
        int tn = w + j * 8;
        brow2[j] = W2tm + (long)(tn * 16 + l15) * 1024 + ncol0 + hl * 2;
      }
      const float* a2 = &d1[l15 * D1LD + hl * 2];
      for (int k = 0; k < 512; k += 4) {
        v2f a = *(const v2f*)(a2 + k);
#pragma unroll
        for (int j = 0; j < 4; j++) {
          v2f b = *(const v2f*)(brow2[j] + k);
          dec[j] = wmma4(a, b, dec[j]);
        }
      }
    }
    __syncthreads();
  }
  // epilogue: imputation select + masked recon accumulation
  float rloc = 0.0f;
#pragma unroll
  for (int j = 0; j < 4; j++) {
    int tn = w + j * 8;
    int dcol = tn * 16 + l15;
    float bv = b2m[dcol];
#pragma unroll
    for (int i = 0; i < 8; i++) {
      int r = tr * 16 + hl * 8 + i;  // b index
      float dv = dec[j][i] + bv;
      int present = pm[r * MM + m];
      long ei = ((long)r * MM + m) * DD + dcol;
      float ev = emb[ei];
      out_imp[ei] = present ? ev : dv;
      if (present) {
        float df = dv - ev;
        rloc += df * df;
      }
    }
  }
  for (int off = 16; off > 0; off >>= 1) rloc += __shfl_down(rloc, off, 32);
  if (lane == 0) atomicAdd(&recon_acc[m], rloc);
}

// uncertainty head: one wave per (b,m)
__global__ void k_unc(const float* __restrict__ z, const float* __restrict__ W1,
                      const float* __restrict__ b1, const float* __restrict__ W2,
                      const float* __restrict__ b2, const int* __restrict__ pm,
                      float* __restrict__ out_unc) {
  int gw = (blockIdx.x * blockDim.x + threadIdx.x) >> 5;
  int lane = threadIdx.x & 31;
  if (gw >= BB * MM) return;
  int b = gw / MM, m = gw % MM;
  const float* W1m = W1 + (long)m * LL * 64;
  const float* b1m = b1 + (long)m * 64;
  const float* W2m = W2 + (long)m * 64;
  float acc = 0.0f;
#pragma unroll
  for (int jj = 0; jj < 2; jj++) {
    int j = lane + jj * 32;
    float hv = b1m[j];
    for (int l = 0; l < LL; l++) hv += z[(long)b * LL + l] * W1m[l * 64 + j];
    acc += gelu_f(hv) * W2m[j];
  }
  for (int off = 16; off > 0; off >>= 1) acc += __shfl_down(acc, off, 32);
  if (lane == 0) {
    float u = 1.0f / (1.0f + expf(-(acc + b2[m])));
    out_unc[b * MM + m] = pm[b * MM + m] ? 0.0f : u;
  }
}

// scalar losses
__global__ void k_fin(const float* __restrict__ cnt, const float* __restrict__ acc,
                      float* __restrict__ out_sc) {
  if (threadIdx.x == 0 && blockIdx.x == 0) {
    float kl = -0.5f * acc[0] / (float)BB;
    float hs = 0.0f, rs = 0.0f;
#pragma unroll
    for (int m = 0; m < MM; m++) {
      float c = cnt[m];
      float has = (c > 0.0f) ? 1.0f : 0.0f;
      hs += has;
      rs += has * (acc[1 + m] / (fmaxf(c, 1.0f) * (float)DD));
    }
    out_sc[0] = kl;
    out_sc[1] = rs / fmaxf(hs, 1.0f);
  }
}

// ---------------------------------------------------------------------------
extern "C" void kernel_launch(void* const* d_in, const int* in_sizes, int n_in,
                              void* d_out, int out_size, void* d_ws, size_t ws_size,
                              hipStream_t stream) {
  const float* emb = (const float*)d_in[0];
  const int* pm = (const int*)d_in[1];
  const float* quality = (const float*)d_in[2];
  const float* eps = (const float*)d_in[3];
  // params (dict insertion order)
  const float* encW1[3] = {(const float*)d_in[4], (const float*)d_in[10], (const float*)d_in[16]};
  const float* encB1[3] = {(const float*)d_in[5], (const float*)d_in[11], (const float*)d_in[17]};
  const float* encG[3]  = {(const float*)d_in[6], (const float*)d_in[12], (const float*)d_in[18]};
  const float* encBe[3] = {(const float*)d_in[7], (const float*)d_in[13], (const float*)d_in[19]};
  const float* encW2[3] = {(const float*)d_in[8], (const float*)d_in[14], (const float*)d_in[20]};
  const float* encB2[3] = {(const float*)d_in[9], (const float*)d_in[15], (const float*)d_in[21]};
  const float* Wq = (const float*)d_in[22]; const float* bq = (const float*)d_in[23];
  const float* Wk = (const float*)d_in[24]; const float* bk = (const float*)d_in[25];
  const float* Wv = (const float*)d_in[26]; const float* bv = (const float*)d_in[27];
  const float* Wo = (const float*)d_in[28]; const float* bo = (const float*)d_in[29];
  const float* muW1 = (const float*)d_in[30]; const float* mub1 = (const float*)d_in[31];
  const float* muW2 = (const float*)d_in[32]; const float* mub2 = (const float*)d_in[33];
  const float* lvW1 = (const float*)d_in[34]; const float* lvb1 = (const float*)d_in[35];
  const float* lvW2 = (const float*)d_in[36]; const float* lvb2 = (const float*)d_in[37];
  const float* decW1 = (const float*)d_in[38]; const float* decb1 = (const float*)d_in[39];
  const float* decW2 = (const float*)d_in[40]; const float* decb2 = (const float*)d_in[41];
  const float* uncW1 = (const float*)d_in[42]; const float* uncb1 = (const float*)d_in[43];
  const float* uncW2 = (const float*)d_in[44]; const float* uncb2 = (const float*)d_in[45];
  const float* qtW1 = (const float*)d_in[46]; const float* qtb1 = (const float*)d_in[47];
  const float* qtW2 = (const float*)d_in[48]; const float* qtb2 = (const float*)d_in[49];

  float* out = (float*)d_out;
  float* out_unc = out + (size_t)BB * MM * DD;
  float* out_sc = out_unc + (size_t)BB * MM;

  float* ws = (float*)d_ws;
  const size_t OFF_H = 0;
  const size_t OFF_ENC = OFF_H + (size_t)BB * MM * 512;
  const size_t OFF_Q = OFF_ENC + (size_t)BB * MM * EE;
  const size_t OFF_K = OFF_Q + (size_t)BB * MM * EE;
  const size_t OFF_V = OFF_K + (size_t)BB * MM * EE;
  const size_t OFF_ATT = OFF_V + (size_t)BB * MM * EE;
  const size_t OFF_JNT = OFF_ATT + (size_t)BB * MM * EE;
  const size_t OFF_MU = OFF_JNT + (size_t)BB * EE;
  const size_t OFF_LV = OFF_MU + (size_t)BB * LL;
  const size_t OFF_Z = OFF_LV + (size_t)BB * LL;
  const size_t OFF_QW = OFF_Z + (size_t)BB * LL;
  const size_t OFF_AA = OFF_QW + (size_t)BB * MM;
  const size_t OFF_CC = OFF_AA + (size_t)MM * 512;
  const size_t OFF_SUM = OFF_CC + (size_t)MM * 512;
  const size_t OFF_SSQ = OFF_SUM + (size_t)MM * 512;
  const size_t OFF_CNT = OFF_SSQ + (size_t)MM * 512;
  const size_t OFF_ACC = OFF_CNT + 8;
  // transposed weight scratch
  const size_t OFF_W1T = OFF_ACC + 8;                       // enc W1t (reused/scale)
  const size_t OFF_W2T = OFF_W1T + (size_t)MM * 512 * 512;  // enc W2t (reused/scale)
  const size_t OFF_WQT = OFF_W2T + (size_t)MM * 512 * 64;
  const size_t OFF_WKT = OFF_WQT + (size_t)EE * EE;
  const size_t OFF_WVT = OFF_WKT + (size_t)EE * EE;
  const size_t OFF_WOT = OFF_WVT + (size_t)EE * EE;
  const size_t OFF_MU1T = OFF_WOT + (size_t)EE * EE;
  const size_t OFF_MU2T = OFF_MU1T + (size_t)EE * LL;
  const size_t OFF_LV1T = OFF_MU2T + (size_t)LL * LL;
  const size_t OFF_LV2T = OFF_LV1T + (size_t)EE * LL;
  const size_t OFF_D1T = OFF_LV2T + (size_t)LL * LL;
  const size_t OFF_D2T = OFF_D1T + (size_t)MM * LL * 1024;

  hipMemsetAsync(ws + OFF_CNT, 0, 16 * sizeof(float), stream);
  k_cnt<<<BB / 256, 256, 0, stream>>>(pm, ws + OFF_CNT);
  k_qw<<<BB / 256, 256, 0, stream>>>(quality, qtW1, qtb1, qtW2, qtb2, ws + OFF_QW);

  // transpose the small fixed weights once per call
  k_transpose<<<dim3(49, 1), 256, 0, stream>>>(Wq, ws + OFF_WQT, EE, EE);
  k_transpose<<<dim3(49, 1), 256, 0, stream>>>(Wk, ws + OFF_WKT, EE, EE);
  k_transpose<<<dim3(49, 1), 256, 0, stream>>>(Wv, ws + OFF_WVT, EE, EE);
  k_transpose<<<dim3(49, 1), 256, 0, stream>>>(Wo, ws + OFF_WOT, EE, EE);
  k_transpose<<<dim3(28, 1), 256, 0, stream>>>(muW1, ws + OFF_MU1T, EE, LL);
  k_transpose<<<dim3(16, 1), 256, 0, stream>>>(muW2, ws + OFF_MU2T, LL, LL);
  k_transpose<<<dim3(28, 1), 256, 0, stream>>>(lvW1, ws + OFF_LV1T, EE, LL);
  k_transpose<<<dim3(16, 1), 256, 0, stream>>>(lvW2, ws + OFF_LV2T, LL, LL);
  k_transpose<<<dim3(256, MM), 256, 0, stream>>>(decW1, ws + OFF_D1T, LL, 1024);
  k_transpose<<<dim3(1024, MM), 256, 0, stream>>>(decW2, ws + OFF_D2T, 1024, DD);

  // multi-scale masked encode
  const int offE[3] = {0, 64, 96};
  for (int s = 0; s < 3; s++) {
    int ds = 512 >> s, ls = 64 >> s;
    hipMemsetAsync(ws + OFF_SUM, 0, 2 * MM * 512 * sizeof(float), stream);
    k_transpose<<<dim3(1024, MM), 256, 0, stream>>>(encW1[s], ws + OFF_W1T, DD, ds);
    k_transpose<<<dim3(128, MM), 256, 0, stream>>>(encW2[s], ws + OFF_W2T, ds, ls);
    int groupsh = (BB / 16) * ((ds / 16 + NBLK - 1) / NBLK);
    k_gemm_bias_t<<<dim3(groupsh / 8, MM), 256, 0, stream>>>(
        emb, (long)MM * DD, (long)DD, ws + OFF_W1T, (long)DD * ds, encB1[s],
        (long)ds, ws + OFF_H, (long)MM * ds, (long)ds, BB, DD, ds);
    k_stats<<<dim3((MM * ds + 255) / 256, 64), 256, 0, stream>>>(
        ws + OFF_H, pm, ws + OFF_SUM, ws + OFF_SSQ, ds);
    k_bnfin<<<(MM * ds + 255) / 256, 256, 0, stream>>>(
        ws + OFF_SUM, ws + OFF_SSQ, ws + OFF_CNT, encG[s], encBe[s],
        ws + OFF_AA, ws + OFF_CC, ds);
    k_hn<<<2048, 256, 0, stream>>>(ws + OFF_H, ws + OFF_AA, ws + OFF_CC, ds);
    int groupse = (BB / 16) * ((ls / 16 + NBLK - 1) / NBLK);
    k_gemm_enc_t<<<dim3((groupse + 7) / 8, MM), 256, 0, stream>>>(
        ws + OFF_H, (long)MM * ds, (long)ds, ws + OFF_W2T, (long)ds * ls,
        encB2[s], (long)ls, ws + OFF_QW, pm, ws + OFF_ENC + offE[s],
        (long)MM * EE, (long)EE, BB, ds, ls);
  }

  // attention projections (N=112 -> 7 tiles -> 2 groups of 4/3 per row tile)
  int groupsq = ((BB * MM) / 16) * ((EE / 16 + NBLK - 1) / NBLK);
  k_gemm_bias_t<<<dim3(groupsq / 8, 1), 256, 0, stream>>>(
      ws + OFF_ENC, (long)EE, 0L, ws + OFF_WQT, 0L, bq, 0L, ws + OFF_Q, (long)EE,
      0L, BB * MM, EE, EE);
  k_gemm_bias_t<<<dim3(groupsq / 8, 1), 256, 0, stream>>>(
      ws + OFF_ENC, (long)EE, 0L, ws + OFF_WKT, 0L, bk, 0L, ws + OFF_K, (long)EE,
      0L, BB * MM, EE, EE);
  k_gemm_bias_t<<<dim3(groupsq / 8, 1), 256, 0, stream>>>(
      ws + OFF_ENC, (long)EE, 0L, ws + OFF_WVT, 0L, bv, 0L, ws + OFF_V, (long)EE,
      0L, BB * MM, EE, EE);
  k_attn<<<(BB * HH) / 256, 256, 0, stream>>>(ws + OFF_Q, ws + OFF_K, ws + OFF_V,
                                              pm, ws + OFF_ATT);
  // attended @ Wo + bo (reuse Q buffer)
  k_gemm_bias_t<<<dim3(groupsq / 8, 1), 256, 0, stream>>>(
      ws + OFF_ATT, (long)EE, 0L, ws + OFF_WOT, 0L, bo, 0L, ws + OFF_Q, (long)EE,
      0L, BB * MM, EE, EE);
  k_pool<<<(BB * EE) / 256, 256, 0, stream>>>(ws + OFF_ENC, ws + OFF_Q, pm,
                                              ws + OFF_JNT);

  // variational heads
  k_head<<<BB / 16, 128, 0, stream>>>(ws + OFF_JNT, ws + OFF_MU1T, mub1,
                                      ws + OFF_MU2T, mub2, ws + OFF_MU);
  k_head<<<BB / 16, 128, 0, stream>>>(ws + OFF_JNT, ws + OFF_LV1T, lvb1,
                                      ws + OFF_LV2T, lvb2, ws + OFF_LV);
  k_z<<<(BB * LL) / 256, 256, 0, stream>>>(ws + OFF_MU, ws + OFF_LV, eps,
                                           ws + OFF_Z, ws + OFF_ACC);

  // fused decode + imputation + recon
  k_decode<<<dim3(BB / 16, MM), 256, 0, stream>>>(ws + OFF_Z, ws + OFF_D1T, decb1,
                                                  ws + OFF_D2T, decb2, emb, pm,
                                                  out, ws + OFF_ACC + 1);
  // uncertainty head
  k_unc<<<(BB * MM * 32) / 256, 256, 0, stream>>>(ws + OFF_Z, uncW1, uncb1,
                                                  uncW2, uncb2, pm, out_unc);
  // scalar losses
  k_fin<<<1, 32, 0, stream>>>(ws + OFF_CNT, ws + OFF_ACC, out_sc);
}